// NestedWindowAverageFeatExtractor_12678743458380
// MI455X (gfx1250) — compile-verified
//
#include <hip/hip_runtime.h>
#include <cstdint>

// NestedWindowAverageFeatExtractor for MI455X (gfx1250).
// out[b, s, y, x, c] = avg_pool_same(in, k_s)[b,y,x,c] - in[b,y,x,c], k in {7,15,31}.
// Fused tile kernel: TDM tensor_load_to_lds (interior tiles) /
// global_load_async_to_lds_b32 (border tiles) -> LDS summed-area table ->
// 3 window deviations, non-temporal stores.

typedef int v4i __attribute__((ext_vector_type(4)));
typedef int v8i __attribute__((ext_vector_type(8)));

#define B_   16
#define H_   512
#define W_   512
#define C_   3
#define T_   32            // output tile edge
#define R_   15            // max radius (k=31)
#define P_   62            // loaded tile edge in pixels: T_ + 2*R_
#define RW_  186           // floats per LDS row: P_*C_ (packed; matches TDM row packing)
#define NS_  3
#define ROWF (W_ * C_)     // 1536 floats per image row

__global__ __launch_bounds__(256)
void nwafe_kernel(const float* __restrict__ in, float* __restrict__ out) {
  __shared__ float tile[P_][RW_];     // 62*186*4 = 46128 B

  const int tid = threadIdx.x;
  const int x0 = blockIdx.x * T_;
  const int y0 = blockIdx.y * T_;
  const int b  = blockIdx.z;
  const int oX = x0 - R_, oY = y0 - R_;          // tile origin incl. halo
  const size_t inBase = (size_t)b * H_ * W_ * C_;

  const bool interior = (blockIdx.x >= 1) & (blockIdx.x <= 14) &
                        (blockIdx.y >= 1) & (blockIdx.y <= 14);

  if (interior) {
    // ---- One TDM descriptor loads the whole 62x186-float patch into LDS.
    if (tid < 32) {  // wave 0 issues the DMA; TDM ignores EXEC
      const uint32_t ldsBase = (uint32_t)(uintptr_t)&tile[0][0];
      const uint64_t gaddr   = (uint64_t)(uintptr_t)
          (in + inBase + ((size_t)oY * W_ + oX) * C_);
      v4i g0; v8i g1; v4i gz;
      // D# group 0: count=1 (valid), lds_addr, global_addr[56:0], type=2
      g0[0] = 1;
      g0[1] = (int)ldsBase;
      g0[2] = (int)(uint32_t)gaddr;
      g0[3] = (int)((((uint32_t)(gaddr >> 32)) & 0x01FFFFFFu) | (2u << 30));
      // D# group 1: wg_mask=0, data_size=2 (4B), no barrier/iterate/pad,
      // tensor_dim0=1536, tensor_dim1=512, tile_dim0=186, tile_dim1=62,
      // tile_dim2=0, tensor_dim0_stride=1536, tensor_dim1_stride=0
      g1[0] = (2 << 16);
      g1[1] = (int)((1536u & 0xFFFFu) << 16);                   // dim0 lo16
      g1[2] = (int)((1536u >> 16) | ((512u & 0xFFFFu) << 16));  // dim0 hi | dim1 lo
      g1[3] = (int)((512u >> 16) | ((uint32_t)RW_ << 16));      // dim1 hi | tile_dim0
      g1[4] = P_;                                               // tile_dim1 | tile_dim2=0
      g1[5] = ROWF;                                             // dim0_stride lo32
      g1[6] = 0;                                                // dim0_stride hi | dim1_stride lo
      g1[7] = 0;
      gz[0] = gz[1] = gz[2] = gz[3] = 0;                        // groups 2/3: unused dims
      asm volatile("tensor_load_to_lds %0, %1, %2, %3"
                   :: "s"(g0), "s"(g1), "s"(gz), "s"(gz) : "memory");
      __builtin_amdgcn_s_wait_tensorcnt(0);
    }
  } else {
    // ---- Border tiles: per-lane async global->LDS, zero out-of-image cells
    // (TF 'SAME' pooling excludes padding; zeros + count division is exact).
    const int NCELL = P_ * RW_;                   // 11532
    for (int i = tid; i < NCELL; i += 256) {
      int yy = i / RW_;
      int j  = i - yy * RW_;                      // j = 3*xpix + c (contiguous)
      int gy = oY + yy;
      int xp = oX + j / C_;
      uint32_t ldsOff = (uint32_t)(uintptr_t)&tile[yy][j];
      if (gy >= 0 && gy < H_ && xp >= 0 && xp < W_) {
        long long idx = ((long long)gy * W_ + oX) * C_ + j;     // >= 0 when valid
        uint64_t gaddr = (uint64_t)(uintptr_t)(in + inBase + (size_t)idx);
        asm volatile("global_load_async_to_lds_b32 %0, %1, off"
                     :: "v"(ldsOff), "v"(gaddr) : "memory");
      } else {
        tile[yy][j] = 0.0f;
      }
    }
    asm volatile("s_wait_asynccnt 0" ::: "memory");
  }
  __syncthreads();

  // ---- Inclusive prefix along x: one thread per (row, channel); stride-3 walk.
  if (tid < P_ * C_) {
    int yy = tid / C_;
    int c  = tid - yy * C_;
    float acc = 0.0f;
#pragma unroll 1
    for (int xx = 0; xx < P_; ++xx) {
      acc += tile[yy][xx * C_ + c];
      tile[yy][xx * C_ + c] = acc;
    }
  }
  __syncthreads();

  // ---- Inclusive prefix along y: one thread per packed column j.
  if (tid < RW_) {
    int j = tid;
    float acc = 0.0f;
#pragma unroll 1
    for (int yy = 0; yy < P_; ++yy) {
      acc += tile[yy][j];
      tile[yy][j] = acc;
    }
  }
  __syncthreads();

  // ---- Three window deviations per output element; SAT boundary (-1) -> 0.
  auto satv = [&](int a, int bp, int c) -> float {
    int ia = a  < 0 ? 0 : a;
    int ib = bp < 0 ? 0 : bp;
    float v = tile[ia][ib * C_ + c];
    return (a < 0 || bp < 0) ? 0.0f : v;
  };

  const int NOUT = T_ * T_ * C_;                  // 3072
  for (int i = tid; i < NOUT; i += 256) {
    int ty  = i / (T_ * C_);
    int rem = i - ty * (T_ * C_);
    int tx  = rem / C_;
    int c   = rem - tx * C_;
    int gy = y0 + ty, gx = x0 + tx;
    float xval = in[inBase + ((size_t)gy * W_ + gx) * C_ + c];
    const int radii[NS_] = {3, 7, 15};
#pragma unroll
    for (int s = 0; s < NS_; ++s) {
      int r   = radii[s];
      int yLo = max(gy - r, 0), yHi = min(gy + r, H_ - 1);
      int xLo = max(gx - r, 0), xHi = min(gx + r, W_ - 1);
      int yl = yLo - oY - 1, yh = yHi - oY;       // in [-1, 61]
      int xl = xLo - oX - 1, xh = xHi - oX;
      float sum = satv(yh, xh, c) - satv(yl, xh, c)
                - satv(yh, xl, c) + satv(yl, xl, c);
      float cnt = (float)((yHi - yLo + 1) * (xHi - xLo + 1));
      float dev = sum / cnt - xval;
      // Write-once 144MB stream: non-temporal so it doesn't evict the
      // fully-L2-resident 48MB input.
      float* op = out + (((((size_t)b * NS_ + s) * H_) + gy) * W_ + gx) * C_ + c;
      __builtin_nontemporal_store(dev, op);
    }
  }
}

extern "C" void kernel_launch(void* const* d_in, const int* in_sizes, int n_in,
                              void* d_out, int out_size, void* d_ws, size_t ws_size,
                              hipStream_t stream) {
  (void)in_sizes; (void)n_in; (void)out_size; (void)d_ws; (void)ws_size;
  const float* in = (const float*)d_in[0];
  float* out = (float*)d_out;
  dim3 grid(W_ / T_, H_ / T_, B_);   // 16 x 16 x 16 tiles
  dim3 block(256, 1, 1);             // 8 waves (wave32)
  nwafe_kernel<<<grid, block, 0, stream>>>(in, out);
}